// Filter_56581899157926
// MI455X (gfx1250) — compile-verified
//
#include <hip/hip_runtime.h>
#include <stdint.h>

// Problem constants (match reference).
constexpr int   kB       = 8;
constexpr int   kN       = 4096;
constexpr int   kC       = 8;
constexpr int   kMaxDet  = 100;
constexpr int   kThreads = 1024;            // 32 wave32s per workgroup
constexpr int   kPer     = kN / kThreads;   // 4 boxes per thread, in registers
constexpr float kScoreT  = 0.05f;
constexpr float kNmsT    = 0.5f;
constexpr float kNeg     = -1000000000.0f;

typedef __attribute__((ext_vector_type(4))) unsigned int u32x4;
typedef __attribute__((ext_vector_type(8))) int          i32x8;
typedef __attribute__((ext_vector_type(4))) int          i32x4;

// Use the CDNA5 Tensor Data Mover when the toolchain exposes it.
#if defined(__AMDGCN__) && defined(__has_builtin)
#if __has_builtin(__builtin_amdgcn_tensor_load_to_lds) && \
    __has_builtin(__builtin_amdgcn_s_wait_tensorcnt)
#define USE_TDM 1
#endif
#endif

// Monotone map: float ordering -> unsigned ordering (strict order preserved).
__device__ __forceinline__ unsigned monoKey(float f) {
  unsigned b = __float_as_uint(f);
  return (b & 0x80000000u) ? ~b : (b | 0x80000000u);
}

__global__ __launch_bounds__(kThreads) void nms3d_kernel(
    const float* __restrict__ boxes, const float* __restrict__ probs,
    float* __restrict__ out) {
  // 96 KB of the WGP's 320 KB LDS: the batch's boxes (for box_j broadcast + gather).
  __shared__ float              sBox[kN * 6];
  __shared__ unsigned long long sCell[2];        // ping-pong argmax cells
  __shared__ int                sKeepIdx[kMaxDet];
  __shared__ int                sKeepVal[kMaxDet];

  const int b   = blockIdx.x;
  const int tid = threadIdx.x;
  const float* gBox = boxes + (size_t)b * kN * 6;
  const float* gPrb = probs + (size_t)b * kN * kC;

#ifdef USE_TDM
  // Wave 0 kicks off an async TDM copy of this batch's boxes (96 KB) into LDS.
  // 2-D descriptor: one row of 24576 f32 elements (tile == tensor).
  if (tid < 32) {
    unsigned long long ga = (unsigned long long)(uintptr_t)gBox;
    unsigned ldsAddr = (unsigned)(uintptr_t)(void*)&sBox[0];
    u32x4 g0;
    g0[0] = 1u;                                          // count=1, user descriptor
    g0[1] = ldsAddr;                                     // lds_addr [63:32]
    g0[2] = (unsigned)(ga & 0xFFFFFFFFull);              // global_addr lo
    g0[3] = (unsigned)((ga >> 32) & 0x1FFFFFFull)        // global_addr [56:32]
            | (2u << 30);                                // type = 2 ("image")
    i32x8 g1;
    g1[0] = (int)(2u << 16);                             // data_size = 4B
    g1[1] = (int)(24576u << 16);                         // tensor_dim0 lo16
    g1[2] = (int)(1u << 16);                             // dim0 hi=0; tensor_dim1=1
    g1[3] = (int)(24576u << 16);                         // dim1 hi=0; tile_dim0=24576
    g1[4] = 1;                                           // tile_dim1=1, tile_dim2=0
    g1[5] = 24576;                                       // tensor_dim0_stride lo32
    g1[6] = (int)(24576u << 16);                         // stride0 hi=0; stride1 lo16
    g1[7] = 0;
    i32x4 z4 = {0, 0, 0, 0};
#if __clang_major__ >= 23
    i32x8 z8 = {0, 0, 0, 0, 0, 0, 0, 0};
    __builtin_amdgcn_tensor_load_to_lds(g0, g1, z4, z4, z8, 0);
#else
    __builtin_amdgcn_tensor_load_to_lds(g0, g1, z4, z4, 0);
#endif
  }
#endif

  // Each thread owns 4 boxes entirely in VGPRs; suppression never re-reads memory.
  float lo[kPer][3], hi[kPer][3], vol[kPer], sw[kPer];
#pragma unroll
  for (int k = 0; k < kPer; ++k) {
    const int i = tid + k * kThreads;
    const float* bp = gBox + (size_t)i * 6;
    float v = 1.0f;
#pragma unroll
    for (int d = 0; d < 3; ++d) {
      lo[k][d] = bp[d];
      hi[k][d] = bp[3 + d];
      v *= fmaxf(hi[k][d] - lo[k][d], 0.0f);
    }
    vol[k] = v;
    const float* pp = gPrb + (size_t)i * kC;
    float s = pp[0];
    int lab = 0;
#pragma unroll
    for (int c = 1; c < kC; ++c) {         // strict > keeps first max (jnp.argmax)
      float p = pp[c];
      if (p > s) { s = p; lab = c; }
    }
    sw[k] = (s > kScoreT && lab != 0) ? s : kNeg;
#ifndef USE_TDM
#pragma unroll
    for (int d = 0; d < 3; ++d) {
      sBox[i * 6 + d]     = lo[k][d];
      sBox[i * 6 + 3 + d] = hi[k][d];
    }
#endif
  }

  if (tid == 0) { sCell[0] = 0ull; sCell[1] = 0ull; }
#ifdef USE_TDM
  if (tid < 32) __builtin_amdgcn_s_wait_tensorcnt(0);   // TDM done before LDS reads
#endif
  __syncthreads();

  const unsigned goodThresh = monoKey(kNeg * 0.5f);     // good <=> s > NEG*0.5

  for (int it = 0; it < kMaxDet; ++it) {
    const int p = it & 1;
    // Pack (value, first-index) into one monotone u64 key: max => exact argmax
    // with jnp's lowest-index tie-break (low word = N-1-i).
    unsigned long long key = 0ull;
#pragma unroll
    for (int k = 0; k < kPer; ++k) {
      const int i = tid + k * kThreads;
      unsigned long long kk =
          ((unsigned long long)monoKey(sw[k]) << 32) | (unsigned)(kN - 1 - i);
      key = (kk > key) ? kk : key;
    }
    // wave32 butterfly reduction (5 steps), then one LDS u64 atomic per wave.
#pragma unroll
    for (int off = 16; off > 0; off >>= 1) {
      unsigned long long o = __shfl_xor(key, off, 32);
      key = (o > key) ? o : key;
    }
    if ((tid & 31) == 0) atomicMax(&sCell[p], key);
    __syncthreads();

    const unsigned long long best = sCell[p];
    const int  j    = (kN - 1) - (int)(best & 0xFFFFFFFFull);
    const bool good = (unsigned)(best >> 32) > goodThresh;

    // Broadcast selected box from LDS (bank-broadcast, no conflicts).
    const float jl0 = sBox[j * 6 + 0], jl1 = sBox[j * 6 + 1], jl2 = sBox[j * 6 + 2];
    const float jh0 = sBox[j * 6 + 3], jh1 = sBox[j * 6 + 4], jh2 = sBox[j * 6 + 5];
    const float jvol = fmaxf(jh0 - jl0, 0.f) * fmaxf(jh1 - jl1, 0.f) *
                       fmaxf(jh2 - jl2, 0.f);

    if (tid == 0) {
      sKeepIdx[it]  = j;
      sKeepVal[it]  = good ? 1 : 0;
      sCell[p ^ 1]  = 0ull;            // reset the *other* cell for next iteration
    }

    if (good) {
#pragma unroll
      for (int k = 0; k < kPer; ++k) {
        const int i = tid + k * kThreads;
        const float ix = fmaxf(fminf(hi[k][0], jh0) - fmaxf(lo[k][0], jl0), 0.f);
        const float iy = fmaxf(fminf(hi[k][1], jh1) - fmaxf(lo[k][1], jl1), 0.f);
        const float iz = fmaxf(fminf(hi[k][2], jh2) - fmaxf(lo[k][2], jl2), 0.f);
        const float inter = ix * iy * iz;
        const float iou   = inter / (vol[k] + jvol - inter + 1e-8f);
        if (iou > kNmsT || i == j) sw[k] = kNeg;
      }
    }
    __syncthreads();   // orders cell reset before next iteration's atomics
  }

  // Gather outputs: d_out = [boxes (B*100*6) f32 | scores (B*100) f32 | labels (B*100) i32]
  if (tid < kMaxDet) {
    const int idx = sKeepIdx[tid];
    const int val = sKeepVal[tid];
    float* ob = out + ((size_t)b * kMaxDet + tid) * 6;
#pragma unroll
    for (int d = 0; d < 6; ++d) ob[d] = val ? sBox[idx * 6 + d] : 0.0f;
    // Recompute score/label for the kept index (cheaper than 32 KB more LDS).
    const float* pp = gPrb + (size_t)idx * kC;
    float s = pp[0];
    int lab = 0;
#pragma unroll
    for (int c = 1; c < kC; ++c) {
      float pv = pp[c];
      if (pv > s) { s = pv; lab = c; }
    }
    out[kB * kMaxDet * 6 + b * kMaxDet + tid] = val ? s : -1.0f;
    int* outi = (int*)out;
    outi[kB * kMaxDet * 6 + kB * kMaxDet + b * kMaxDet + tid] = val ? lab : -1;
  }
}

extern "C" void kernel_launch(void* const* d_in, const int* in_sizes, int n_in,
                              void* d_out, int out_size, void* d_ws, size_t ws_size,
                              hipStream_t stream) {
  const float* boxes = (const float*)d_in[0];   // (B, N, 6) f32
  const float* probs = (const float*)d_in[1];   // (B, N, C) f32
  float* out = (float*)d_out;                   // 6400 elements, see layout above
  nms3d_kernel<<<dim3(kB), dim3(kThreads), 0, stream>>>(boxes, probs, out);
}